// AttentionalGNN_75436805587526
// MI455X (gfx1250) — compile-verified
//
#include <hip/hip_runtime.h>

typedef float v2f __attribute__((ext_vector_type(2)));
typedef float v8f __attribute__((ext_vector_type(8)));

#define USE_ASYNC_LDS 1

// ---------------------------------------------------------------------------
// Generic strided batched GEMM (f32) on V_WMMA_F32_16X16X4_F32, LDS-tiled.
//   C[z][m][n] = alpha * sum_k A[z][m][k]*B[z][k][n]  (+ bias[m]) (+ R[z][m][n])
// z = z0*g1 + z1 with independent per-z0/z1 base strides per operand.
// Block = 256 threads (8 waves); block tile 64(M) x 128(N); K-block 16.
// Waves arranged 2(m) x 4(n); each wave computes a 32x32 tile (2x2 WMMA tiles).
// Panels staged to LDS via async global->LDS loads (ASYNCcnt), fragments read
// as ds_load_b64 (A[64][18]dw, B transposed [128][18]dw, pad kills conflicts).
// Requires: M%64==0, Ncols%128==0, K%16==0.
//
// WMMA f32 16x16x4 fragment layouts (CDNA5 ISA 7.12.2):
//   A 16x4 : lane%16 = row m, K = vgpr + 2*(lane>=16)
//   B 4x16 : lane%16 = col n, K = vgpr + 2*(lane>=16)
//   C 16x16: lane%16 = col n, row m = vgpr + 8*(lane>=16)
// ---------------------------------------------------------------------------
__global__ __launch_bounds__(256) void gemm_f32_wmma(
    const float* __restrict__ A, const float* __restrict__ B,
    float* __restrict__ C, const float* __restrict__ bias,
    const float* __restrict__ R, int K,
    long lda_m, long lda_k, long ldb_k, long ldb_n,
    long ldc_m, long ldc_n, long ldr_m, long ldr_n,
    int g1,
    long Ag0, long Ag1, long Bg0, long Bg1,
    long Cg0, long Cg1, long Rg0, long Rg1,
    float alpha)
{
  __shared__ v2f As2[64 * 9];    // A panel: [row 0..63][k 0..15], pitch 18 dwords
  __shared__ v2f Bs2[128 * 9];   // B panel transposed: [n 0..127][k 0..15]

  const int t    = threadIdx.x;
  const int lane = t & 31;
  const int wave = t >> 5;
  const int wm   = wave >> 2;          // 0..1
  const int wn   = wave & 3;           // 0..3
  const int hi   = lane >> 4;          // half-wave selector
  const int l16  = lane & 15;

  const int z  = blockIdx.z;
  const int z0 = z / g1;
  const int z1 = z - z0 * g1;

  const int mBase = blockIdx.y << 6;   // 64-row block tile
  const int nBase = blockIdx.x << 7;   // 128-col block tile

  // ---- staging assignments -------------------------------------------------
  // A: 64x16 elems / 256 thr = 4 each: row = t&63, k0 = (t>>6)*4
  const int arow = t & 63;
  const int ak0  = (t >> 6) << 2;
  const float* aptr = A + (long)z0 * Ag0 + (long)z1 * Ag1
                        + (long)(mBase + arow) * lda_m + (long)ak0 * lda_k;
  // B: 16x128 elems / 256 thr = 8 each: n = t&127, k0 = (t>>7)*8
  const int bn  = t & 127;
  const int bk0 = (t >> 7) << 3;
  const float* bptr = B + (long)z0 * Bg0 + (long)z1 * Bg1
                        + (long)(nBase + bn) * ldb_n + (long)bk0 * ldb_k;

  const unsigned aLdsBase = (unsigned)(uintptr_t)(&As2[0]) + arow * 72u + ak0 * 4u;
  const unsigned bLdsBase = (unsigned)(uintptr_t)(&Bs2[0]) + bn * 72u + bk0 * 4u;

  v8f acc00 = {}, acc01 = {}, acc10 = {}, acc11 = {};

  const v2f* __restrict__ Arow0 = &As2[(wm * 32 + l16) * 9];
  const v2f* __restrict__ Arow1 = &As2[(wm * 32 + 16 + l16) * 9];
  const v2f* __restrict__ Brow0 = &Bs2[(wn * 32 + l16) * 9];
  const v2f* __restrict__ Brow1 = &Bs2[(wn * 32 + 16 + l16) * 9];

  for (int kb = 0; kb < K; kb += 16) {
#if USE_ASYNC_LDS
    // ---- async global -> LDS staging (ASYNCcnt path) -----------------------
#pragma unroll
    for (int j = 0; j < 4; ++j) {
      const float* g = aptr + (long)j * lda_k;
      unsigned l = aLdsBase + (unsigned)(j * 4);
      asm volatile("global_load_async_to_lds_b32 %0, %1, off"
                   :: "v"(l), "v"(g) : "memory");
    }
#pragma unroll
    for (int j = 0; j < 8; ++j) {
      const float* g = bptr + (long)j * ldb_k;
      unsigned l = bLdsBase + (unsigned)(j * 4);
      asm volatile("global_load_async_to_lds_b32 %0, %1, off"
                   :: "v"(l), "v"(g) : "memory");
    }
    asm volatile("s_wait_asynccnt 0x0" ::: "memory");
#else
    // ---- fallback: regular loads + ds stores -------------------------------
    {
      float* asf = (float*)&As2[0];
      float* bsf = (float*)&Bs2[0];
#pragma unroll
      for (int j = 0; j < 4; ++j)
        asf[arow * 18 + ak0 + j] = aptr[(long)j * lda_k];
#pragma unroll
      for (int j = 0; j < 8; ++j)
        bsf[bn * 18 + bk0 + j] = bptr[(long)j * ldb_k];
    }
#endif
    aptr += 16 * lda_k;
    bptr += 16 * ldb_k;
    __syncthreads();

    // ---- compute: 4 K-steps of 4, 2x2 tiles per wave -----------------------
#pragma unroll
    for (int ks = 0; ks < 4; ++ks) {
      const int ci = (ks << 1) + hi;   // (kk>>1) + half  -> b64 slot
      v2f a0 = Arow0[ci];
      v2f a1 = Arow1[ci];
      v2f b0 = Brow0[ci];
      v2f b1 = Brow1[ci];
      acc00 = __builtin_amdgcn_wmma_f32_16x16x4_f32(false, a0, false, b0, (short)0, acc00, false, false);
      acc01 = __builtin_amdgcn_wmma_f32_16x16x4_f32(false, a0, false, b1, (short)0, acc01, false, false);
      acc10 = __builtin_amdgcn_wmma_f32_16x16x4_f32(false, a1, false, b0, (short)0, acc10, false, false);
      acc11 = __builtin_amdgcn_wmma_f32_16x16x4_f32(false, a1, false, b1, (short)0, acc11, false, false);
    }
    __syncthreads();
  }

  // ---- epilogue ------------------------------------------------------------
  float* Cp = C + (long)z0 * Cg0 + (long)z1 * Cg1;
  const float* Rp = R ? (R + (long)z0 * Rg0 + (long)z1 * Rg1) : nullptr;

  v8f* accs[2][2] = {{&acc00, &acc01}, {&acc10, &acc11}};
#pragma unroll
  for (int mt = 0; mt < 2; ++mt) {
#pragma unroll
    for (int nt = 0; nt < 2; ++nt) {
      v8f acc = *accs[mt][nt];
#pragma unroll
      for (int r = 0; r < 8; ++r) {
        const int m = mBase + wm * 32 + mt * 16 + r + (hi << 3);
        const int n = nBase + wn * 32 + nt * 16 + l16;
        float v = acc[r] * alpha;
        if (bias) v += bias[m];
        if (Rp)   v += Rp[(long)m * ldr_m + (long)n * ldr_n];
        Cp[(long)m * ldc_m + (long)n * ldc_n] = v;
      }
    }
  }
}

// ---------------------------------------------------------------------------
// In-place row softmax: one block per row of length M.
// ---------------------------------------------------------------------------
__global__ __launch_bounds__(256) void softmax_rows_kernel(float* __restrict__ S, int M)
{
  __shared__ float red[8];
  __shared__ float bval;
  const int tid = threadIdx.x;
  float* p = S + (long)blockIdx.x * (long)M;

  float mx = -3.4e38f;
  for (int j = tid; j < M; j += 256) mx = fmaxf(mx, p[j]);
#pragma unroll
  for (int o = 16; o; o >>= 1) mx = fmaxf(mx, __shfl_xor(mx, o, 32));
  if ((tid & 31) == 0) red[tid >> 5] = mx;
  __syncthreads();
  if (tid == 0) {
    float m2 = red[0];
    for (int w = 1; w < 8; ++w) m2 = fmaxf(m2, red[w]);
    bval = m2;
  }
  __syncthreads();
  mx = bval;

  float s = 0.f;
  for (int j = tid; j < M; j += 256) {
    float e = expf(p[j] - mx);
    p[j] = e;
    s += e;
  }
  __syncthreads();
#pragma unroll
  for (int o = 16; o; o >>= 1) s += __shfl_xor(s, o, 32);
  if ((tid & 31) == 0) red[tid >> 5] = s;
  __syncthreads();
  if (tid == 0) {
    float t = 0.f;
    for (int w = 0; w < 8; ++w) t += red[w];
    bval = t;
  }
  __syncthreads();
  const float inv = 1.0f / bval;
  for (int j = tid; j < M; j += 256) p[j] *= inv;
}

// ---------------------------------------------------------------------------
// BatchNorm1d (eval) + ReLU, in place. Layout [B, C2, N].
// ---------------------------------------------------------------------------
__global__ __launch_bounds__(256) void bn_relu_kernel(
    float* __restrict__ H, const float* __restrict__ gamma,
    const float* __restrict__ beta, const float* __restrict__ rm,
    const float* __restrict__ rv, int C2, long N, long total)
{
  long i = (long)blockIdx.x * 256 + threadIdx.x;
  if (i >= total) return;
  const int c = (int)((i / N) % C2);
  const float sc = gamma[c] / sqrtf(rv[c] + 1e-5f);
  const float v = (H[i] - rm[c]) * sc + beta[c];
  H[i] = fmaxf(v, 0.f);
}

// ---------------------------------------------------------------------------

extern "C" void kernel_launch(void* const* d_in, const int* in_sizes, int n_in,
                              void* d_out, int out_size, void* d_ws, size_t ws_size,
                              hipStream_t stream)
{
  const float* desc0 = (const float*)d_in[0];
  const float* desc1 = (const float*)d_in[1];
  const float* Wq = (const float*)d_in[2];
  const float* bq = (const float*)d_in[3];
  const float* Wk = (const float*)d_in[4];
  const float* bk = (const float*)d_in[5];
  const float* Wv = (const float*)d_in[6];
  const float* bv = (const float*)d_in[7];
  const float* Wm = (const float*)d_in[8];
  const float* bm = (const float*)d_in[9];
  const float* W1 = (const float*)d_in[10];
  const float* b1 = (const float*)d_in[11];
  const float* gamma = (const float*)d_in[12];
  const float* beta  = (const float*)d_in[13];
  const float* rm = (const float*)d_in[14];
  const float* rv = (const float*)d_in[15];
  const float* W2 = (const float*)d_in[16];
  const float* b2 = (const float*)d_in[17];

  const int Bb = 2, D = 256, Nn = 1024, Hh = 4, Lnum = 18, D2 = 512;
  const long act = (long)Bb * D * Nn;           // 524288 floats

  float* ws = (float*)d_ws;
  float* X[2][2];                                // ping-pong d0/d1
  X[0][0] = ws + 0 * act;  X[0][1] = ws + 1 * act;
  X[1][0] = ws + 2 * act;  X[1][1] = ws + 3 * act;
  float* Q    = ws + 4 * act;                    // also reused as attn-out
  float* Kb   = ws + 5 * act;                    // also reused as msg
  float* V    = ws + 6 * act;
  float* Hbuf = ws + 7 * act;                    // 2*act (B x 2D x N)
  float* S    = ws + 9 * act;                    // 16*act (B x H x N x N)
  // total workspace need: 25*act floats = 50 MiB

  auto gemm = [&](const float* A, const float* Bm, float* Cm, const float* bias,
                  const float* R, int M, int Ncols, int K,
                  long lda_m, long lda_k, long ldb_k, long ldb_n,
                  long ldc_m, long ldc_n, long ldr_m, long ldr_n,
                  int g0, int g1,
                  long Ag0, long Ag1, long Bg0, long Bg1,
                  long Cg0, long Cg1, long Rg0, long Rg1, float alpha) {
    dim3 grid(Ncols / 128, M / 64, g0 * g1);
    gemm_f32_wmma<<<grid, dim3(256), 0, stream>>>(
        A, Bm, Cm, bias, R, K, lda_m, lda_k, ldb_k, ldb_n,
        ldc_m, ldc_n, ldr_m, ldr_n, g1,
        Ag0, Ag1, Bg0, Bg1, Cg0, Cg1, Rg0, Rg1, alpha);
  };

  const float* cur0 = desc0;
  const float* cur1 = desc1;

  for (int i = 0; i < Lnum; ++i) {
    const bool cross = (i & 1);                  // ['self','cross'] * 9
    float* out0 = (i == Lnum - 1) ? (float*)d_out        : X[i & 1][0];
    float* out1 = (i == Lnum - 1) ? (float*)d_out + act  : X[i & 1][1];

    const float* Wq_i = Wq + (long)i * D * D;   const float* bq_i = bq + (long)i * D;
    const float* Wk_i = Wk + (long)i * D * D;   const float* bk_i = bk + (long)i * D;
    const float* Wv_i = Wv + (long)i * D * D;   const float* bv_i = bv + (long)i * D;
    const float* Wm_i = Wm + (long)i * D * D;   const float* bm_i = bm + (long)i * D;
    const float* W1_i = W1 + (long)i * D2 * D2; const float* b1_i = b1 + (long)i * D2;
    const float* W2_i = W2 + (long)i * D * D2;  const float* b2_i = b2 + (long)i * D;
    const float* g_i  = gamma + (long)i * D2;   const float* be_i = beta + (long)i * D2;
    const float* rm_i = rm + (long)i * D2;      const float* rv_i = rv + (long)i * D2;

    for (int s = 0; s < 2; ++s) {
      const float* x   = (s == 0) ? cur0 : cur1;
      const float* src = cross ? ((s == 0) ? cur1 : cur0) : x;
      float* outx = (s == 0) ? out0 : out1;

      // q = Wq x + bq ; k = Wk src + bk ; v = Wv src + bv   (conv1x1)
      gemm(Wq_i, x,   Q,  bq_i, nullptr, D, Nn, D, D, 1, Nn, 1, Nn, 1, 0, 0,
           Bb, 1, 0, 0, (long)D * Nn, 0, (long)D * Nn, 0, 0, 0, 1.f);
      gemm(Wk_i, src, Kb, bk_i, nullptr, D, Nn, D, D, 1, Nn, 1, Nn, 1, 0, 0,
           Bb, 1, 0, 0, (long)D * Nn, 0, (long)D * Nn, 0, 0, 0, 1.f);
      gemm(Wv_i, src, V,  bv_i, nullptr, D, Nn, D, D, 1, Nn, 1, Nn, 1, 0, 0,
           Bb, 1, 0, 0, (long)D * Nn, 0, (long)D * Nn, 0, 0, 0, 1.f);

      // scores[b,h,n,m] = (1/8) sum_d q[b,d,h,n] k[b,d,h,m]   (dh=64)
      gemm(Q, Kb, S, nullptr, nullptr, Nn, Nn, 64,
           1, (long)Hh * Nn,               // A=q: row(n) stride 1, k(d) stride H*N
           (long)Hh * Nn, 1,               // B=k: k(d) stride H*N, col(m) stride 1
           Nn, 1, 0, 0,
           Bb, Hh,
           (long)D * Nn, Nn, (long)D * Nn, Nn,
           (long)Hh * Nn * Nn, (long)Nn * Nn, 0, 0, 0.125f);

      softmax_rows_kernel<<<Bb * Hh * Nn, 256, 0, stream>>>(S, Nn);

      // out[b,d,h,n] = sum_m prob[b,h,n,m] v[b,d,h,m]   (reuse Q as attn-out)
      gemm(V, S, Q, nullptr, nullptr, 64, Nn, Nn,
           (long)Hh * Nn, 1,               // A=v: row(d) stride H*N, k(m) stride 1
           1, Nn,                          // B=prob^T: k(m) stride 1, col(n) stride N
           (long)Hh * Nn, 1, 0, 0,
           Bb, Hh,
           (long)D * Nn, Nn,
           (long)Hh * Nn * Nn, (long)Nn * Nn,
           (long)D * Nn, Nn, 0, 0, 1.f);

      // msg = Wm attn-out + bm   (reuse Kb as msg)
      gemm(Wm_i, Q, Kb, bm_i, nullptr, D, Nn, D, D, 1, Nn, 1, Nn, 1, 0, 0,
           Bb, 1, 0, 0, (long)D * Nn, 0, (long)D * Nn, 0, 0, 0, 1.f);

      // h = W1[:, :D] x + b1 ;  h += W1[:, D:] msg   (split-K of concat)
      gemm(W1_i, x, Hbuf, b1_i, nullptr, D2, Nn, D, D2, 1, Nn, 1, Nn, 1, 0, 0,
           Bb, 1, 0, 0, (long)D * Nn, 0, (long)D2 * Nn, 0, 0, 0, 1.f);
      gemm(W1_i + D, Kb, Hbuf, nullptr, Hbuf, D2, Nn, D, D2, 1, Nn, 1, Nn, 1, Nn, 1,
           Bb, 1, 0, 0, (long)D * Nn, 0, (long)D2 * Nn, 0, (long)D2 * Nn, 0, 1.f);

      // BN(eval) + ReLU in place
      const long tot = (long)Bb * D2 * Nn;
      bn_relu_kernel<<<(int)((tot + 255) / 256), 256, 0, stream>>>(
          Hbuf, g_i, be_i, rm_i, rv_i, D2, (long)Nn, tot);

      // new_x = x + (W2 h + b2)
      gemm(W2_i, Hbuf, outx, b2_i, x, D, Nn, D2, D2, 1, Nn, 1, Nn, 1, Nn, 1,
           Bb, 1, 0, 0, (long)D2 * Nn, 0, (long)D * Nn, 0, (long)D * Nn, 0, 1.f);
    }

    cur0 = out0;
    cur1 = out1;
  }
}